// SparseCloudConvolution_67173288509589
// MI455X (gfx1250) — compile-verified
//
#include <hip/hip_runtime.h>
#include <hip/hip_bf16.h>

typedef __attribute__((ext_vector_type(2))) float v2f;
typedef __attribute__((ext_vector_type(8))) float v8f;

// ---------------------------------------------------------------------------
// Kernel 1: Y[i, tl, f] = sum_c X[i,c] * K[t0+tl, c, f]
// One wave computes a 16-row x 64-col tile for one term t, using
// V_WMMA_F32_16X16X4_F32 (full fp32 matrix pipeline, K=4 per instruction).
//
// A-frag (16x4 f32, 2 VGPR):  lane l -> row m=l&15, holds K = 2*(l>>4)+{0,1}
// B-frag (4x16 f32, 2 VGPR):  lane l -> col n=l&15, holds K = 2*(l>>4)+{0,1}
// C/D   (16x16 f32, 8 VGPR):  vgpr j -> row j + 8*(l>>4), col n=l&15
// ---------------------------------------------------------------------------
__global__ __launch_bounds__(128) void scc_gemm_wmma(
    const float* __restrict__ X,    // (N_IN, C)
    const float* __restrict__ Kmat, // (T, C, F)
    float* __restrict__ Y,          // (N_IN, tc*F) chunk workspace
    int N_IN, int C, int F, int t0, int tc)
{
    const int lane    = threadIdx.x & 31;
    const int wave    = threadIdx.x >> 5;
    const int nTiles  = (N_IN + 15) >> 4;
    const int tileIdx = blockIdx.x * 4 + wave;
    if (tileIdx >= nTiles) return;

    const int tl = blockIdx.y;          // local term index within chunk
    const int t  = t0 + tl;
    const int r0 = tileIdx << 4;

    const float* __restrict__ Kt = Kmat + (size_t)t * C * F;

    const int m  = lane & 15;
    const int n  = lane & 15;
    const int kb = (lane >> 4) << 1;    // 0 for lanes 0-15, 2 for lanes 16-31

    int rowA = r0 + m;
    if (rowA >= N_IN) rowA = N_IN - 1;  // clamp (N_IN%16==0 in practice)
    const float* __restrict__ Xr = X + (size_t)rowA * C;

    v8f acc0 = (v8f)0.0f, acc1 = (v8f)0.0f, acc2 = (v8f)0.0f, acc3 = (v8f)0.0f;

    for (int k = 0; k < C; k += 4) {
        v2f a;
        a.x = Xr[k + kb + 0];
        a.y = Xr[k + kb + 1];

        const float* __restrict__ Kr0 = Kt + (size_t)(k + kb) * F;
        const float* __restrict__ Kr1 = Kr0 + F;

        v2f b0; b0.x = Kr0[n +  0]; b0.y = Kr1[n +  0];
        v2f b1; b1.x = Kr0[n + 16]; b1.y = Kr1[n + 16];
        v2f b2; b2.x = Kr0[n + 32]; b2.y = Kr1[n + 32];
        v2f b3; b3.x = Kr0[n + 48]; b3.y = Kr1[n + 48];

        acc0 = __builtin_amdgcn_wmma_f32_16x16x4_f32(false, a, false, b0,
                                                     (short)0, acc0, false, false);
        acc1 = __builtin_amdgcn_wmma_f32_16x16x4_f32(false, a, false, b1,
                                                     (short)0, acc1, false, false);
        acc2 = __builtin_amdgcn_wmma_f32_16x16x4_f32(false, a, false, b2,
                                                     (short)0, acc2, false, false);
        acc3 = __builtin_amdgcn_wmma_f32_16x16x4_f32(false, a, false, b3,
                                                     (short)0, acc3, false, false);
    }

    const int ldY = tc * F;
#pragma unroll
    for (int j = 0; j < 8; ++j) {
        const int row = r0 + j + ((lane >> 4) << 3);
        if (row < N_IN) {
            float* __restrict__ Yr = Y + (size_t)row * ldY + (size_t)tl * F;
            Yr[n +  0] = acc0[j];
            Yr[n + 16] = acc1[j];
            Yr[n + 32] = acc2[j];
            Yr[n + 48] = acc3[j];
        }
    }
}

// ---------------------------------------------------------------------------
// Kernel 2: per-edge gather + register t-reduction + 64 atomic adds.
// One wave per edge; each lane owns F/32 output features.
//   out[o, f] += sum_tl EF[t0+tl, e] * Y[i, tl, f]
// Y rows are 2KB contiguous -> fully coalesced 128B wave requests, L2-resident.
// ---------------------------------------------------------------------------
__global__ __launch_bounds__(256) void scc_edge_scatter(
    const float* __restrict__ Y,        // (N_IN, tc*F)
    const float* __restrict__ EF,       // (T, E)
    const long long* __restrict__ IDX,  // (E, 2) int64 [out, in]
    float* __restrict__ OUT,            // (N_OUT, F)
    int E, int F, int t0, int tc)
{
    const int lane = threadIdx.x & 31;
    const int eid  = blockIdx.x * (blockDim.x >> 5) + (threadIdx.x >> 5);
    if (eid >= E) return;

    const int o = (int)IDX[2 * (size_t)eid + 0];
    const int i = (int)IDX[2 * (size_t)eid + 1];

    const float* __restrict__ Yr = Y + (size_t)i * ((size_t)tc * F);

    const int FQ = F >> 5;                // features per lane (2 for F=64)
    float acc[4] = {0.0f, 0.0f, 0.0f, 0.0f};

    for (int tl = 0; tl < tc; ++tl) {
        const float w = EF[(size_t)(t0 + tl) * E + eid];   // broadcast load
        const float* __restrict__ Yt = Yr + (size_t)tl * F;
        for (int q = 0; q < FQ && q < 4; ++q)
            acc[q] = fmaf(w, Yt[(q << 5) + lane], acc[q]);
    }

    float* __restrict__ Or = OUT + (size_t)o * F;
    for (int q = 0; q < FQ && q < 4; ++q)
        atomicAdd(&Or[(q << 5) + lane], acc[q]);
}

// ---------------------------------------------------------------------------
// Kernel 3: initialize output with bias (d_out is poisoned before timing).
// ---------------------------------------------------------------------------
__global__ void scc_init_out(float* __restrict__ OUT,
                             const float* __restrict__ bias,
                             int total, int F)
{
    const int idx = blockIdx.x * blockDim.x + threadIdx.x;
    if (idx < total) OUT[idx] = bias[idx % F];
}

extern "C" void kernel_launch(void* const* d_in, const int* in_sizes, int n_in,
                              void* d_out, int out_size, void* d_ws, size_t ws_size,
                              hipStream_t stream) {
    const float*     X    = (const float*)d_in[0];      // node_features (N_IN, C)
    const float*     EF   = (const float*)d_in[1];      // edge_features (T, E)
    const long long* IDX  = (const long long*)d_in[2];  // indices (E, 2) int64
    const float*     Km   = (const float*)d_in[3];      // kernel (T, C, F)
    const float*     bias = (const float*)d_in[4];      // bias (F,)
    // d_in[5] is the out_size scalar on device; N_OUT derived from out_size/F.

    const int F    = in_sizes[4];
    const int E    = in_sizes[2] / 2;
    const int T    = in_sizes[1] / E;
    const int C    = in_sizes[3] / (T * F);
    const int N_IN = in_sizes[0] / C;

    float* OUT = (float*)d_out;
    float* Yws = (float*)d_ws;

    // out = bias (broadcast)
    scc_init_out<<<(out_size + 255) / 256, 256, 0, stream>>>(OUT, bias, out_size, F);

    // Chunk t-dimension to fit workspace: Y chunk = N_IN * TC * F floats.
    const size_t bytesPerT = (size_t)N_IN * (size_t)F * sizeof(float);
    int TC = (bytesPerT > 0) ? (int)(ws_size / bytesPerT) : 1;
    if (TC > T) TC = T;
    if (TC < 1) TC = 1;

    const int nTiles = (N_IN + 15) / 16;

    for (int t0 = 0; t0 < T; t0 += TC) {
        const int tc = (T - t0 < TC) ? (T - t0) : TC;

        // Stage 1: Y[:, tl, :] = X @ K[t0+tl]  (fp32 WMMA)
        dim3 gGrid((nTiles + 3) / 4, tc);
        scc_gemm_wmma<<<gGrid, 128, 0, stream>>>(X, Km, Yws, N_IN, C, F, t0, tc);

        // Stage 2: edge gather + t-reduction + atomic scatter into OUT
        const int wavesPerBlock = 8;
        dim3 eGrid((E + wavesPerBlock - 1) / wavesPerBlock);
        scc_edge_scatter<<<eGrid, wavesPerBlock * 32, 0, stream>>>(
            Yws, EF, IDX, OUT, E, F, t0, tc);
    }
}